// GIN_11879879544634
// MI455X (gfx1250) — compile-verified
//
#include <hip/hip_runtime.h>

typedef float v2f __attribute__((ext_vector_type(2)));
typedef float v8f __attribute__((ext_vector_type(8)));

#define N_NODES 50000
#define N_EDGES 1600000
#define HID     128
#define N_CLASS 10
#define N_GRAPH 128
#define BN_EPS  1e-5f

// ---------------------------------------------------------------- zero
__global__ void zero_f32(float* __restrict__ p, int n4) {
    int i = blockIdx.x * 256 + threadIdx.x;
    if (i < n4) ((float4*)p)[i] = float4{0.f, 0.f, 0.f, 0.f};
}

// ---------------------------------------------------------------- scatter-add: agg[dst] += x[src]
// wave-per-edge: lane f handles features f, f+32, f+64, f+96 (coalesced 128B)
__global__ __launch_bounds__(256) void gin_scatter(const float* __restrict__ x,
                                                   const int* __restrict__ ei,
                                                   float* __restrict__ agg) {
    int wave = threadIdx.x >> 5;
    int lane = threadIdx.x & 31;
    int e = blockIdx.x * 8 + wave;
    if (e >= N_EDGES) return;
    int s = ei[e];
    int d = ei[N_EDGES + e];
    const float* xr = x + (size_t)s * HID;
    float*       ar = agg + (size_t)d * HID;
#pragma unroll
    for (int j = 0; j < 4; ++j)
        atomicAdd(ar + lane + 32 * j, xr[lane + 32 * j]);
}

// ---------------------------------------------------------------- BN fold: scale/shift (b2 folded in)
__global__ void bn_prep(const float* __restrict__ gamma, const float* __restrict__ beta,
                        const float* __restrict__ rmean, const float* __restrict__ rvar,
                        const float* __restrict__ b2, float* __restrict__ sc,
                        float* __restrict__ sh) {
    int c = threadIdx.x;
    if (c < HID) {
        float s = gamma[c] * rsqrtf(rvar[c] + BN_EPS);
        sc[c] = s;
        sh[c] = (b2[c] - rmean[c]) * s + beta[c];
    }
}

// ---------------------------------------------------------------- fused GEMM (fp32 WMMA 16x16x4)
// out[N,128] = epilogue( (A0 (+A1)) @ W + ... )
//   ADD_AGG: A = A0 + A1      BN: out = relu(acc*sc + sh)   else: out = relu(acc + bias)
// Block: 256 thr = 8 waves covering a 128x128 tile.
//   wave -> 32 rows x 64 cols: row group (wave&3)*32, col group (wave>>2)*64.
//   Each B fragment feeds TWO WMMAs (both row stripes) -> 4 WMMA per ds_load_2addr_b64.
// W staged in LDS transposed: Wt[k2*80 + n] = {W[2k2][n], W[2k2+1][n]}; pad 80 keeps the
// two half-waves (k-rows offset by 2) on disjoint bank halves -> conflict-free ds_load_b64.
// Explicit B/A double-buffering + (256,1) VGPR headroom keep next-step loads in flight.
template <bool ADD_AGG, bool BN>
__global__ __launch_bounds__(256, 1) void gin_gemm(const float* __restrict__ A0,
                                                   const float* __restrict__ A1,
                                                   const float* __restrict__ W,
                                                   const float* __restrict__ bias,
                                                   const float* __restrict__ sc,
                                                   const float* __restrict__ sh,
                                                   float* __restrict__ out) {
    __shared__ v2f Wt[64 * 80];  // 40,960 B

    for (int idx = threadIdx.x; idx < 64 * 128; idx += 256) {
        int n  = idx & 127;
        int k2 = idx >> 7;
        v2f w;
        w.x = W[(2 * k2) * HID + n];
        w.y = W[(2 * k2 + 1) * HID + n];
        Wt[k2 * 80 + n] = w;
    }
    __syncthreads();

    int wave = threadIdx.x >> 5;
    int lane = threadIdx.x & 31;
    int mgrp = wave & 3;   // 4 row groups of 32 rows
    int ngrp = wave >> 2;  // 2 col groups of 64 cols
    int m0   = blockIdx.x * 128 + mgrp * 32;

    int rowS0 = m0 + (lane & 15);       // A stripe 0 (rows m0..m0+15)
    int rowS1 = rowS0 + 16;             // A stripe 1 (rows m0+16..m0+31)
    if (rowS0 >= N_NODES) rowS0 = N_NODES - 1;  // clamp loads; stores are guarded
    if (rowS1 >= N_NODES) rowS1 = N_NODES - 1;
    int khalf = lane >> 4;              // 0: K=0,1   1: K=2,3 (A-frag layout)

    const float* a0s0 = A0 + (size_t)rowS0 * HID + khalf * 2;
    const float* a0s1 = A0 + (size_t)rowS1 * HID + khalf * 2;
    const float* a1s0 = ADD_AGG ? (A1 + (size_t)rowS0 * HID + khalf * 2) : nullptr;
    const float* a1s1 = ADD_AGG ? (A1 + (size_t)rowS1 * HID + khalf * 2) : nullptr;

    int colBase = ngrp * 64 + (lane & 15);
    const v2f* wrow = Wt + khalf * 80 + colBase;  // + k2*2*80 per k-step

    // acc[s*4 + nt]: stripe s (0/1) x col tile nt (0..3)
    v8f acc[8];
#pragma unroll
    for (int t = 0; t < 8; ++t) acc[t] = (v8f){0.f, 0.f, 0.f, 0.f, 0.f, 0.f, 0.f, 0.f};

    // prologue loads (k-step 0)
    v2f b[4];
#pragma unroll
    for (int nt = 0; nt < 4; ++nt) b[nt] = wrow[nt * 16];
    v2f aS0 = *(const v2f*)(a0s0);
    v2f aS1 = *(const v2f*)(a0s1);
    if (ADD_AGG) {
        v2f g0 = *(const v2f*)(a1s0);
        v2f g1 = *(const v2f*)(a1s1);
        aS0.x += g0.x; aS0.y += g0.y;
        aS1.x += g1.x; aS1.y += g1.y;
    }

#pragma unroll
    for (int kk = 0; kk < HID; kk += 4) {
        // prefetch k-step kk+4 (B from LDS, A from global) before this step's WMMAs
        v2f bn[4];
        v2f a0n = aS0, a1n = aS1;
        if (kk + 4 < HID) {
#pragma unroll
            for (int nt = 0; nt < 4; ++nt) bn[nt] = wrow[((kk + 4) >> 1) * 80 + nt * 16];
            a0n = *(const v2f*)(a0s0 + kk + 4);
            a1n = *(const v2f*)(a0s1 + kk + 4);
            if (ADD_AGG) {
                v2f g0 = *(const v2f*)(a1s0 + kk + 4);
                v2f g1 = *(const v2f*)(a1s1 + kk + 4);
                a0n.x += g0.x; a0n.y += g0.y;
                a1n.x += g1.x; a1n.y += g1.y;
            }
        } else {
#pragma unroll
            for (int nt = 0; nt < 4; ++nt) bn[nt] = b[nt];
        }

#pragma unroll
        for (int nt = 0; nt < 4; ++nt) {
            acc[nt] = __builtin_amdgcn_wmma_f32_16x16x4_f32(
                false, aS0, false, b[nt], (short)0, acc[nt], false, false);
            acc[4 + nt] = __builtin_amdgcn_wmma_f32_16x16x4_f32(
                false, aS1, false, b[nt], (short)0, acc[4 + nt], false, false);
        }
#pragma unroll
        for (int nt = 0; nt < 4; ++nt) b[nt] = bn[nt];
        aS0 = a0n;
        aS1 = a1n;
    }

    // epilogue + store (D layout: VGPR v -> M = v (lanes 0-15) / 8+v (lanes 16-31), N = lane&15)
    int rhalf = (lane >> 4) << 3;
#pragma unroll
    for (int s = 0; s < 2; ++s) {
        int rbase = m0 + s * 16 + rhalf;
#pragma unroll
        for (int nt = 0; nt < 4; ++nt) {
            int col = colBase + nt * 16;
            float p0 = BN ? sc[col] : bias[col];
            float p1 = BN ? sh[col] : 0.f;
#pragma unroll
            for (int v = 0; v < 8; ++v) {
                int row = rbase + v;
                float val = acc[s * 4 + nt][v];
                val = BN ? fmaf(val, p0, p1) : (val + p0);
                val = fmaxf(val, 0.f);
                if (row < N_NODES) out[(size_t)row * HID + col] = val;
            }
        }
    }
}

// ---------------------------------------------------------------- mean-pool accumulation
__global__ __launch_bounds__(256) void gin_pool(const float* __restrict__ x,
                                                const int* __restrict__ batch,
                                                float* __restrict__ pooled,
                                                float* __restrict__ counts) {
    int wave = threadIdx.x >> 5;
    int lane = threadIdx.x & 31;
    int i = blockIdx.x * 8 + wave;
    if (i >= N_NODES) return;
    int g = batch[i];
    const float* xr = x + (size_t)i * HID;
    float*       pr = pooled + (size_t)g * HID;
#pragma unroll
    for (int j = 0; j < 4; ++j)
        atomicAdd(pr + lane + 32 * j, xr[lane + 32 * j]);
    if (lane == 0) atomicAdd(counts + g, 1.0f);
}

// ---------------------------------------------------------------- head: fc + log_softmax
__global__ void gin_head(const float* __restrict__ pooled, const float* __restrict__ counts,
                         const float* __restrict__ fcw, const float* __restrict__ fcb,
                         float* __restrict__ out) {
    int g = threadIdx.x;
    if (g >= N_GRAPH) return;
    float inv = 1.f / fmaxf(counts[g], 1.0f);
    float logits[N_CLASS];
#pragma unroll
    for (int c = 0; c < N_CLASS; ++c) logits[c] = fcb[c];
    for (int k = 0; k < HID; ++k) {
        float p = pooled[(size_t)g * HID + k] * inv;
#pragma unroll
        for (int c = 0; c < N_CLASS; ++c) logits[c] = fmaf(p, fcw[k * N_CLASS + c], logits[c]);
    }
    float m = logits[0];
#pragma unroll
    for (int c = 1; c < N_CLASS; ++c) m = fmaxf(m, logits[c]);
    float s = 0.f;
#pragma unroll
    for (int c = 0; c < N_CLASS; ++c) s += __expf(logits[c] - m);
    float lse = __logf(s);
#pragma unroll
    for (int c = 0; c < N_CLASS; ++c) out[g * N_CLASS + c] = logits[c] - m - lse;
}

// ---------------------------------------------------------------- launch
extern "C" void kernel_launch(void* const* d_in, const int* in_sizes, int n_in,
                              void* d_out, int out_size, void* d_ws, size_t ws_size,
                              hipStream_t stream) {
    const float* x     = (const float*)d_in[0];
    const int*   ei    = (const int*)d_in[1];
    const int*   batch = (const int*)d_in[2];
    const float* W1    = (const float*)d_in[3];
    const float* b1    = (const float*)d_in[4];
    const float* W2    = (const float*)d_in[5];
    const float* b2    = (const float*)d_in[6];
    const float* gamma = (const float*)d_in[7];
    const float* beta  = (const float*)d_in[8];
    const float* rmean = (const float*)d_in[9];
    const float* rvar  = (const float*)d_in[10];
    const float* fcw   = (const float*)d_in[11];
    const float* fcb   = (const float*)d_in[12];
    float* out = (float*)d_out;

    const size_t NF = (size_t)N_NODES * HID;
    float* agg    = (float*)d_ws;
    float* bufT   = agg + NF;
    float* bufX   = bufT + NF;
    float* pooled = bufX + NF;
    float* counts = pooled + (size_t)N_GRAPH * HID;
    float* bnsc   = counts + N_GRAPH;
    float* bnsh   = bnsc + HID;

    const int gemmGrid = (N_NODES + 127) / 128;  // 391
    const float* xin = x;
    for (int l = 0; l < 3; ++l) {
        zero_f32<<<(int)(NF / 4 + 255) / 256, 256, 0, stream>>>(agg, (int)(NF / 4));
        gin_scatter<<<N_EDGES / 8, 256, 0, stream>>>(xin, ei, agg);
        gin_gemm<true, false><<<gemmGrid, 256, 0, stream>>>(
            xin, agg, W1 + (size_t)l * HID * HID, b1 + l * HID, nullptr, nullptr, bufT);
        bn_prep<<<1, 128, 0, stream>>>(gamma + l * HID, beta + l * HID, rmean + l * HID,
                                       rvar + l * HID, b2 + l * HID, bnsc, bnsh);
        gin_gemm<false, true><<<gemmGrid, 256, 0, stream>>>(
            bufT, nullptr, W2 + (size_t)l * HID * HID, nullptr, bnsc, bnsh, bufX);
        xin = bufX;
    }

    int poolN4 = (N_GRAPH * HID + N_GRAPH) / 4;  // pooled + counts contiguous
    zero_f32<<<(poolN4 + 255) / 256, 256, 0, stream>>>(pooled, poolN4);
    gin_pool<<<N_NODES / 8, 256, 0, stream>>>(xin, batch, pooled, counts);
    gin_head<<<1, 128, 0, stream>>>(pooled, counts, fcw, fcb, out);
}